// DynamicMemoryCell_32332513804587
// MI455X (gfx1250) — compile-verified
//
#include <hip/hip_runtime.h>

#define D 4096          // BLOCK_DIM
#define J 128           // N_BLOCKS
#define BK 64           // k-panel width staged in LDS per iteration
#define LDSS (BK + 4)   // padded row stride (words): banks = 4*row + col -> conflict-free
#define BIAS 0.01f

typedef float v2f __attribute__((ext_vector_type(2)));
typedef float v8f __attribute__((ext_vector_type(8)));

// ---------------------------------------------------------------------------
// Async global -> LDS copy of one float4 (per lane), tracked by ASYNCcnt.
// ISA 15.18.3: GLOBAL_LOAD_ASYNC_TO_LDS_B128  vdst(LDS addr), vaddr(64b), off
// Low 32 bits of a generic pointer to LDS are the wave-relative LDS offset.
// ---------------------------------------------------------------------------
__device__ __forceinline__ void async_f4_to_lds(const float* gsrc, float* ldst) {
    unsigned lofs = (unsigned)(reinterpret_cast<size_t>(ldst));
    asm volatile("global_load_async_to_lds_b128 %0, %1, off"
                 :: "v"(lofs), "v"(gsrc)
                 : "memory");
}

__device__ __forceinline__ void wait_async0() {
    asm volatile("s_wait_asynccnt 0x0" ::: "memory");
}

// ---------------------------------------------------------------------------
// Kernel 1: gate g[j] = sigmoid( (hb_j + keys_j) . s )
// ---------------------------------------------------------------------------
__global__ __launch_bounds__(256) void k_gate(const float* __restrict__ h,
                                              const float* __restrict__ keys,
                                              const float* __restrict__ s,
                                              float* __restrict__ g) {
    __shared__ float red[256];
    const int j = blockIdx.x;
    const float* hr = h    + (size_t)j * D;
    const float* kr = keys + (size_t)j * D;
    float acc = 0.0f;
    for (int i = threadIdx.x; i < D; i += 256)
        acc += (hr[i] + kr[i]) * s[i];
    red[threadIdx.x] = acc;
    __syncthreads();
    for (int off = 128; off > 0; off >>= 1) {
        if (threadIdx.x < off) red[threadIdx.x] += red[threadIdx.x + off];
        __syncthreads();
    }
    if (threadIdx.x == 0) {
        float x = red[0];
        g[j] = 1.0f / (1.0f + __expf(-x));
    }
}

// ---------------------------------------------------------------------------
// Kernel 2: main fused WMMA GEMM (fp32, V_WMMA_F32_16X16X4_F32).
//   P[m,j] = sum_k U[m,k]*hb[j,k] + V[m,k]*keys[j,k] + W[m,k]*s[k]
//   out[j,m] = hb[j,m] + g[j] * prelu(P[m,j] + BIAS)
// Grid: 256 blocks (16-row m-tiles) x 8 waves (16-col j-tiles, 8*16 = J).
// U/V/W panels stream HBM -> LDS via async DMA, double-buffered.
// ---------------------------------------------------------------------------
__global__ __launch_bounds__(256) void k_gemm(const float* __restrict__ U,
                                              const float* __restrict__ V,
                                              const float* __restrict__ W,
                                              const float* __restrict__ h,
                                              const float* __restrict__ keys,
                                              const float* __restrict__ s,
                                              const float* __restrict__ g,
                                              const float* __restrict__ prelu_a,
                                              float* __restrict__ out) {
    __shared__ __align__(16) float panel[2][3][16 * LDSS];   // ~26 KB of 320 KB

    const int tid   = threadIdx.x;
    const int wave  = tid >> 5;          // 0..7  -> j-tile
    const int lane  = tid & 31;
    const int m0    = blockIdx.x * 16;   // 256 m-tiles cover D=4096

    const int lrow  = lane & 15;                  // A row (M) / B column (N)
    const int khalf = (lane < 16) ? 0 : 2;        // K pair selected by lane half
    const int jrow  = wave * 16 + lrow;           // this lane's output column j

    const float* __restrict__ hbRow = h    + (size_t)jrow * D;
    const float* __restrict__ keRow = keys + (size_t)jrow * D;

    v8f acc  = {};   // U*hb + V*keys contribution
    v8f accW = {};   // W*s broadcast contribution (same across j)

    // Cooperative async-copy indexing: 256 threads x 1 float4 = 16 x BK panel
    const int    row   = tid >> 4;                     // 0..15
    const int    c4    = (tid & 15) << 2;              // 0,4,...,60
    const size_t gbase = (size_t)(m0 + row) * D + c4;  // + kb at issue time
    const int    lbase = row * LDSS + c4;

    // Prologue: start panel 0 into buffer 0
    async_f4_to_lds(U + gbase, &panel[0][0][lbase]);
    async_f4_to_lds(V + gbase, &panel[0][1][lbase]);
    async_f4_to_lds(W + gbase, &panel[0][2][lbase]);

    int p = 0;
    for (int kb = 0; kb < D; kb += BK) {
        wait_async0();       // my async writes into panel[p] are in LDS
        __syncthreads();     // everyone's writes done; panel[p^1] free again

        if (kb + BK < D) {   // prefetch next panel into the other buffer
            const size_t gn = gbase + (size_t)(kb + BK);
            async_f4_to_lds(U + gn, &panel[p ^ 1][0][lbase]);
            async_f4_to_lds(V + gn, &panel[p ^ 1][1][lbase]);
            async_f4_to_lds(W + gn, &panel[p ^ 1][2][lbase]);
        }

        const float* pU = &panel[p][0][lrow * LDSS];
        const float* pV = &panel[p][1][lrow * LDSS];
        const float* pW = &panel[p][2][lrow * LDSS];

#pragma unroll
        for (int kk = 0; kk < BK; kk += 4) {
            const int lc = kk + khalf;          // LDS A-frag base (K, K+1)
            const int gk = kb + kk + khalf;     // global B-frag base

            v2f aU = *(const v2f*)(pU + lc);
            v2f aV = *(const v2f*)(pV + lc);
            v2f aW = *(const v2f*)(pW + lc);
            v2f bU = *(const v2f*)(hbRow + gk);
            v2f bV = *(const v2f*)(keRow + gk);
            v2f bW = *(const v2f*)(s     + gk);

            acc  = __builtin_amdgcn_wmma_f32_16x16x4_f32(false, aU, false, bU,
                                                         (short)0, acc,  false, false);
            acc  = __builtin_amdgcn_wmma_f32_16x16x4_f32(false, aV, false, bV,
                                                         (short)0, acc,  false, false);
            accW = __builtin_amdgcn_wmma_f32_16x16x4_f32(false, aW, false, bW,
                                                         (short)0, accW, false, false);
        }
        p ^= 1;
    }

    // Epilogue: bias + PReLU + gated residual. D layout: VGPR r holds
    // M = m0 + r (lanes 0-15) / m0 + r + 8 (lanes 16-31), N = lane&15.
    const float a_slope = prelu_a[0];
    const float gj      = g[jrow];
    const int   moff    = (lane < 16) ? 0 : 8;
#pragma unroll
    for (int r = 0; r < 8; ++r) {
        const int m = m0 + r + moff;
        float pre   = acc[r] + accW[r] + BIAS;
        float hsq   = (pre >= 0.0f) ? pre : a_slope * pre;
        float t     = hbRow[m] + gj * hsq;
        out[(size_t)jrow * D + m] = t;
    }
}

// ---------------------------------------------------------------------------
// Kernel 3: inv[j] = rsqrt( sum_m out[j,m]^2 )  (deterministic tree reduce)
// ---------------------------------------------------------------------------
__global__ __launch_bounds__(256) void k_sumsq(const float* __restrict__ out,
                                               float* __restrict__ inv) {
    __shared__ float red[256];
    const int j = blockIdx.x;
    const float* r0 = out + (size_t)j * D;
    float acc = 0.0f;
    for (int i = threadIdx.x; i < D; i += 256) {
        float v = r0[i];
        acc += v * v;
    }
    red[threadIdx.x] = acc;
    __syncthreads();
    for (int off = 128; off > 0; off >>= 1) {
        if (threadIdx.x < off) red[threadIdx.x] += red[threadIdx.x + off];
        __syncthreads();
    }
    if (threadIdx.x == 0) inv[j] = rsqrtf(red[0]);
}

// ---------------------------------------------------------------------------
// Kernel 4: out[j,:] *= inv[j]
// ---------------------------------------------------------------------------
__global__ __launch_bounds__(256) void k_scale(float* __restrict__ out,
                                               const float* __restrict__ inv) {
    const int j = blockIdx.x;
    const float sc = inv[j];
    float* r0 = out + (size_t)j * D;
    for (int i = threadIdx.x; i < D; i += 256) r0[i] *= sc;
}

// ---------------------------------------------------------------------------
extern "C" void kernel_launch(void* const* d_in, const int* in_sizes, int n_in,
                              void* d_out, int out_size, void* d_ws, size_t ws_size,
                              hipStream_t stream) {
    // setup_inputs order: s, h, keys, U, V, W, prelu_a  (all float32)
    const float* s       = (const float*)d_in[0];
    const float* h       = (const float*)d_in[1];
    const float* keys    = (const float*)d_in[2];
    const float* U       = (const float*)d_in[3];
    const float* V       = (const float*)d_in[4];
    const float* W       = (const float*)d_in[5];
    const float* prelu_a = (const float*)d_in[6];
    float* out = (float*)d_out;

    float* ws_g   = (float*)d_ws;        // [J]
    float* ws_inv = (float*)d_ws + J;    // [J]

    k_gate <<<J,      256, 0, stream>>>(h, keys, s, ws_g);
    k_gemm <<<D / 16, 256, 0, stream>>>(U, V, W, h, keys, s, ws_g, prelu_a, out);
    k_sumsq<<<J,      256, 0, stream>>>(out, ws_inv);
    k_scale<<<J,      256, 0, stream>>>(out, ws_inv);
    (void)in_sizes; (void)n_in; (void)out_size; (void)ws_size;
}